// ConstructKernel3d_91156385890464
// MI455X (gfx1250) — compile-verified
//
#include <hip/hip_runtime.h>

typedef __attribute__((ext_vector_type(2))) float v2f;
typedef __attribute__((ext_vector_type(8))) float v8f;

#define OC   128
#define ICG  128
#define KC   16
#define NPAIR (OC * ICG)      // 16384
#define OUTV  343             // 7*7*7 outputs per (o,c) pair
#define LDSTR 344             // padded LDS stride per wave slab

#define WAVES_PER_BLOCK 8
#define PAIRS_PER_WAVE  4

// Trilinear axis weight: (x==f)*(1-r) + (x==f+1)*r  (branches -> v_cndmask)
__device__ __forceinline__ float axw(float x, float f, float r) {
    return (x == f) ? (1.0f - r) : ((x == f + 1.0f) ? r : 0.0f);
}

__global__ __launch_bounds__(WAVES_PER_BLOCK * 32)
void trilinear_construct_wmma(const float* __restrict__ W,
                              const float* __restrict__ P,
                              float* __restrict__ out) {
    __shared__ float lds[WAVES_PER_BLOCK][LDSTR];

    const int tid  = threadIdx.x;
    const int wave = tid >> 5;
    const int lane = tid & 31;
    const int half = lane >> 4;     // half-wave selects K (+2) for A/B operands
    const int n    = lane & 15;     // N column this lane owns in B/C/D tiles
    const float nf = (float)n;

    // A-operand row decode per M-tile: row m = 16*t + (lane&15); m = l*7 + j.
    float lA[4], jA[4];
    bool  vA[4];
#pragma unroll
    for (int t = 0; t < 4; ++t) {
        int m = 16 * t + n;
        vA[t] = (m < 49);
        lA[t] = (float)(m / 7);
        jA[t] = (float)(m % 7);
    }

    v8f zero;
#pragma unroll
    for (int e = 0; e < 8; ++e) zero[e] = 0.0f;

#pragma unroll 1
    for (int it = 0; it < PAIRS_PER_WAVE; ++it) {
        const int p = (blockIdx.x * WAVES_PER_BLOCK + wave) * PAIRS_PER_WAVE + it;
        const float* wp = W + p * KC;
        const float* p0 = P + 0 * NPAIR * KC + p * KC;  // axis 0 -> l weights
        const float* p1 = P + 1 * NPAIR * KC + p * KC;  // axis 1 -> i weights (C)
        const float* p2 = P + 2 * NPAIR * KC + p * KC;  // axis 2 -> j weights (B)

        // Speculative prefetch of the next pair's tap data (global_prefetch_b8)
        __builtin_prefetch(wp + KC, 0, 0);
        __builtin_prefetch(p0 + KC, 0, 0);
        __builtin_prefetch(p1 + KC, 0, 0);
        __builtin_prefetch(p2 + KC, 0, 0);

        v8f acc[4];
#pragma unroll
        for (int t = 0; t < 4; ++t) acc[t] = zero;

#pragma unroll
        for (int q = 0; q < 4; ++q) {
            // Per-lane tap data for this K-chunk. sub = A/B operand VGPR index;
            // half-wave adds +2 to K per the 32-bit 16x4 / 4x16 WMMA layouts.
            float Wk[2], f0[2], r0[2], f1[2], r1[2], f2[2], r2[2], bb[2];
#pragma unroll
            for (int sub = 0; sub < 2; ++sub) {
                const int k = 4 * q + sub + 2 * half;
                Wk[sub] = wp[k];
                float c0 = p0[k] + 3.0f; f0[sub] = floorf(c0); r0[sub] = c0 - f0[sub];
                float c1 = p1[k] + 3.0f; f1[sub] = floorf(c1); r1[sub] = c1 - f1[sub];
                float c2 = p2[k] + 3.0f; f2[sub] = floorf(c2); r2[sub] = c2 - f2[sub];
                // B operand: M2[k, n] = wC_k[n] (axis 1), zero-padded for n >= 7
                bb[sub] = (n < 7) ? axw(nf, f1[sub], r1[sub]) : 0.0f;
            }
            v2f bv; bv[0] = bb[0]; bv[1] = bb[1];

#pragma unroll
            for (int t = 0; t < 4; ++t) {
                // A operand: M1[m, k] = W_k * wA_k[l] * wB_k[j], zero row-pad
                float a0 = vA[t] ? Wk[0] * axw(lA[t], f0[0], r0[0]) * axw(jA[t], f2[0], r2[0]) : 0.0f;
                float a1 = vA[t] ? Wk[1] * axw(lA[t], f0[1], r0[1]) * axw(jA[t], f2[1], r2[1]) : 0.0f;
                v2f av; av[0] = a0; av[1] = a1;
                acc[t] = __builtin_amdgcn_wmma_f32_16x16x4_f32(
                    /*neg_a=*/false, av, /*neg_b=*/false, bv,
                    /*c_mod=*/(short)0, acc[t], /*reuse_a=*/false, /*reuse_b=*/false);
            }
        }

        // Spill C/D tiles (VGPR r = rows r / r+8 per half-wave) into the wave's
        // LDS slab in output order, then stream 343 floats out coalesced.
        const int rbase = half * 8;
#pragma unroll
        for (int t = 0; t < 4; ++t) {
#pragma unroll
            for (int r = 0; r < 8; ++r) {
                const int m = 16 * t + rbase + r;          // m = l*7 + j
                if (m < 49 && n < 7) lds[wave][m * 7 + n] = acc[t][r];
            }
        }
        __builtin_amdgcn_wave_barrier();   // keep DS program order; HW DS is in-order per wave
        float* op = out + p * OUTV;        // out[o,c,l,j,i] flat = p*343 + m*7 + i
        for (int idx = lane; idx < OUTV; idx += 32) op[idx] = lds[wave][idx];
        __builtin_amdgcn_wave_barrier();   // slab reused next pair
    }
}

extern "C" void kernel_launch(void* const* d_in, const int* in_sizes, int n_in,
                              void* d_out, int out_size, void* d_ws, size_t ws_size,
                              hipStream_t stream) {
    const float* W = (const float*)d_in[0];   // [128,128,16] f32
    const float* P = (const float*)d_in[1];   // [3,128,128,16] f32
    float* out = (float*)d_out;               // [128,128,7,7,7] f32
    (void)in_sizes; (void)n_in; (void)out_size; (void)d_ws; (void)ws_size;

    const int pairs = NPAIR;                                   // 16384
    const int pairs_per_block = WAVES_PER_BLOCK * PAIRS_PER_WAVE;  // 32
    dim3 grid(pairs / pairs_per_block);                        // 512
    dim3 block(WAVES_PER_BLOCK * 32);                          // 256 (8 wave32s)
    trilinear_construct_wmma<<<grid, block, 0, stream>>>(W, P, out);
}